// ActionTensorLoss_42777874268646
// MI455X (gfx1250) — compile-verified
//
#include <hip/hip_runtime.h>
#include <stdint.h>

// Problem constants (from the reference: B=4096, A=2048, F=8)
#define BATCH   4096
#define ACT     2048
#define FEAT    8
#define SAMPLE_STRIDE ((ACT + 1) * FEAT)   // 16392 floats per sample per tensor

#define THREADS 256
#define TILE_V4 256                        // float4s per tile per tensor (4 KB)
#define NSTAGES 4                          // async pipeline depth (per-wave)

typedef float f32x4 __attribute__((ext_vector_type(4)));

// ---------------------------------------------------------------------------
// Issue one tile-chunk: every thread async-copies one 16B line of pred and one
// of targ into its private LDS slot. Non-temporal: single-use stream > L2.
// Index is clamped (not predicated) so EXEC stays uniform for the async issue.
// ---------------------------------------------------------------------------
__device__ __forceinline__
void issue_chunk(const float* pbase, const float* tbase,
                 f32x4 (*sp)[TILE_V4], f32x4 (*st)[TILE_V4],
                 int chunk, int buf, int tid, int nvec) {
    int i = min(chunk * TILE_V4 + tid, nvec - 1);
    uint32_t voff = (uint32_t)i * 16u;
    uint32_t lp = (uint32_t)(uintptr_t)&sp[buf][tid];
    uint32_t lt = (uint32_t)(uintptr_t)&st[buf][tid];
    asm volatile("global_load_async_to_lds_b128 %0, %1, %2 offset:0 th:TH_LOAD_NT"
                 :: "v"(lp), "v"(voff), "s"(pbase) : "memory");
    asm volatile("global_load_async_to_lds_b128 %0, %1, %2 offset:0 th:TH_LOAD_NT"
                 :: "v"(lt), "v"(voff), "s"(tbase) : "memory");
}

// Wait until at most `inflight` chunks (2 async ops each) remain outstanding.
// Immediate must be a literal; branches are scalar-uniform per block.
__device__ __forceinline__ void wait_chunks(int inflight) {
    if (inflight >= 3)      asm volatile("s_wait_asynccnt 0x6" ::: "memory");
    else if (inflight == 2) asm volatile("s_wait_asynccnt 0x4" ::: "memory");
    else if (inflight == 1) asm volatile("s_wait_asynccnt 0x2" ::: "memory");
    else                    asm volatile("s_wait_asynccnt 0x0" ::: "memory");
}

// ---------------------------------------------------------------------------
// Kernel 1: one block per sample. Streams only the masked rows (count*8
// floats) through a 4-deep per-thread-private async global->LDS pipeline.
// No intra-loop barriers: each thread consumes only the slots it loaded, so
// per-wave ASYNCcnt ordering is the only synchronization required.
// ws layout (floats): [0..B)   per-sample masked MSE
//                     [B..2B)  (pred_count - target_count)^2
//                     [2B..3B) valid flag (1.0 / 0.0)
// ---------------------------------------------------------------------------
__global__ __launch_bounds__(THREADS)
void per_sample_kernel(const float* __restrict__ pred,
                       const float* __restrict__ targ,
                       float* __restrict__ ws) {
    __shared__ f32x4 sp[NSTAGES][TILE_V4];   // 16 KB
    __shared__ f32x4 st[NSTAGES][TILE_V4];   // 16 KB
    __shared__ float red[THREADS / 32];

    const int b = blockIdx.x;
    const int tid = threadIdx.x;
    const size_t base = (size_t)b * SAMPLE_STRIDE;

    const float tc = targ[base];     // target count (stored as float)
    const float pc = pred[base];     // predicted count
    const int   count = (int)tc;
    const int   nvec  = count * (FEAT / 4);      // float4s of masked action data

    const float* pbase = pred + base + FEAT;     // 16B-aligned (b*65568 + 32)
    const float* tbase = targ + base + FEAT;

    const int nchunks = (nvec + TILE_V4 - 1) / TILE_V4;
    float acc = 0.0f;

    if (nchunks > 0) {
        // ---- prologue: fill the pipeline ----
        const int pre = min(NSTAGES, nchunks);
        for (int s = 0; s < pre; ++s)
            issue_chunk(pbase, tbase, sp, st, s, s, tid, nvec);

        // ---- steady state: wait-oldest, consume, refill ----
        for (int c = 0; c < nchunks; ++c) {
            const int buf = c & (NSTAGES - 1);
            wait_chunks(min(nchunks - c - 1, NSTAGES - 1));

            int idx = c * TILE_V4 + tid;
            if (idx < nvec) {
                f32x4 p = sp[buf][tid];
                f32x4 t = st[buf][tid];
                f32x4 d = p - t;
                acc = fmaf(d.x, d.x, acc);
                acc = fmaf(d.y, d.y, acc);
                acc = fmaf(d.z, d.z, acc);
                acc = fmaf(d.w, d.w, acc);
            }
            // Slot consumed (FMA depends on the ds_load data) -> safe to reuse.
            int nxt = c + NSTAGES;
            if (nxt < nchunks)
                issue_chunk(pbase, tbase, sp, st, nxt, buf, tid, nvec);
        }
    }

    // ---- wave32 reduction, then cross-wave via LDS (fixed, deterministic) ----
    for (int m = 16; m >= 1; m >>= 1)
        acc += __shfl_xor(acc, m, 32);
    if ((tid & 31) == 0) red[tid >> 5] = acc;
    __syncthreads();

    if (tid == 0) {
        float total = 0.0f;
        #pragma unroll
        for (int w = 0; w < THREADS / 32; ++w) total += red[w];

        float mse = (count > 0) ? total / (float)(count * FEAT) : 0.0f;
        float dc  = pc - tc;
        ws[b]             = mse;
        ws[BATCH + b]     = dc * dc;
        ws[2 * BATCH + b] = (count > 0) ? 1.0f : 0.0f;
    }
}

// ---------------------------------------------------------------------------
// Kernel 2: deterministic final reduction over the 4096 per-sample partials.
// ---------------------------------------------------------------------------
__global__ __launch_bounds__(256)
void finalize_kernel(const float* __restrict__ ws, float* __restrict__ out) {
    __shared__ float s0[256], s1[256], s2[256];
    const int tid = threadIdx.x;
    float a0 = 0.0f, a1 = 0.0f, a2 = 0.0f;
    for (int i = tid; i < BATCH; i += 256) {     // fixed per-thread order
        a0 += ws[i];
        a1 += ws[BATCH + i];
        a2 += ws[2 * BATCH + i];
    }
    s0[tid] = a0; s1[tid] = a1; s2[tid] = a2;
    __syncthreads();
    for (int s = 128; s > 0; s >>= 1) {
        if (tid < s) {
            s0[tid] += s0[tid + s];
            s1[tid] += s1[tid + s];
            s2[tid] += s2[tid + s];
        }
        __syncthreads();
    }
    if (tid == 0) {
        const float W_ACTION_COUNT  = 1.0f;
        const float W_ACTION_TENSOR = 2.0f;
        float mse_sum    = s0[0];
        float count_loss = s1[0] / (float)BATCH;
        float n_valid    = s2[0];
        float action     = mse_sum / fmaxf(n_valid, 1.0f);
        float total = W_ACTION_COUNT * count_loss +
                      ((n_valid > 0.0f) ? W_ACTION_TENSOR * action : 0.0f);
        out[0] = total;
    }
}

// ---------------------------------------------------------------------------
extern "C" void kernel_launch(void* const* d_in, const int* in_sizes, int n_in,
                              void* d_out, int out_size, void* d_ws, size_t ws_size,
                              hipStream_t stream) {
    const float* pred = (const float*)d_in[0];
    const float* targ = (const float*)d_in[1];
    float* ws  = (float*)d_ws;     // needs 3 * 4096 * 4 = 48 KB
    float* out = (float*)d_out;

    per_sample_kernel<<<BATCH, THREADS, 0, stream>>>(pred, targ, ws);
    finalize_kernel<<<1, 256, 0, stream>>>(ws, out);
}